// DeeperGCN_31370441130162
// MI455X (gfx1250) — compile-verified
//
#include <hip/hip_runtime.h>
#include <hip/hip_bf16.h>

// DeeperGCN forward for MI455X (gfx1250, wave32).
// N=100000 nodes, E=600000 edges, D=128 channels, L=3 layers, all f32.
//
// Strategy (memory-bound workload, ~23.3 TB/s HBM, 192MB L2):
//  - scatter-max via signed-int atomicMax on float bit patterns (all msgs > 0,
//    accumulator zero-initialized -> ordering is exact and bit-deterministic)
//  - deterministic two-pass BatchNorm (block partials + serial combine)
//  - GEMM (h2+agg)@W + b (+res) with V_WMMA_F32_16X16X4_F32 (exact f32 path)

typedef __attribute__((ext_vector_type(2))) float v2f;
typedef __attribute__((ext_vector_type(8))) float v8f;

#define D 128
#define EPS_MSG 1e-7f
#define EPS_BN  1e-5f
#define BN_BLOCKS 256

// ---------------------------------------------------------------- zero fill
__global__ void dg_zero(float* __restrict__ p, int n) {
  int i = blockIdx.x * blockDim.x + threadIdx.x;
  int stride = gridDim.x * blockDim.x;
  for (; i < n; i += stride) p[i] = 0.0f;
}

// ------------------------------------------------- edge gather + scatter max
// 8 edges per 256-thread block; each lane handles 4 channels (float4 load).
// msg = relu(t[src]) + eps  ->  atomicMax(int) into agg[dst] (init 0).
__global__ __launch_bounds__(256) void dg_scatter_max(
    const float* __restrict__ t, const int* __restrict__ src,
    const int* __restrict__ dst, float* __restrict__ agg, int E) {
  int lane = threadIdx.x & 31;
  int e = blockIdx.x * 8 + (threadIdx.x >> 5);
  if (e >= E) return;
  int s = src[e];
  int d = dst[e];
  const float4* trow = (const float4*)(t + (size_t)s * D);
  float4 v = trow[lane];
  float m0 = fmaxf(v.x, 0.0f) + EPS_MSG;
  float m1 = fmaxf(v.y, 0.0f) + EPS_MSG;
  float m2 = fmaxf(v.z, 0.0f) + EPS_MSG;
  float m3 = fmaxf(v.w, 0.0f) + EPS_MSG;
  int* arow = (int*)(agg + (size_t)d * D) + lane * 4;
  atomicMax(arow + 0, __float_as_int(m0));
  atomicMax(arow + 1, __float_as_int(m1));
  atomicMax(arow + 2, __float_as_int(m2));
  atomicMax(arow + 3, __float_as_int(m3));
}

// ------------------------------------------------------------- WMMA GEMM
// OUT[r, c] = sum_k (T[r,k] + AGG[r,k]) * W[k,c] + bias[c] (+ RES[r,c])
// One 256-thread block per 16-row tile; wave w (of 8) owns columns [16w,16w+16).
// A tile staged in LDS (stride 132 floats to avoid bank conflicts).
__global__ __launch_bounds__(256) void dg_gemm_wmma(
    const float* __restrict__ T, const float* __restrict__ AGG,
    const float* __restrict__ W, const float* __restrict__ bias,
    const float* __restrict__ RES, float* __restrict__ OUT, int N) {
  __shared__ float As[16 * 132];

  int row0 = blockIdx.x * 16;
  // cooperative load of A = T + AGG  (16 x 128), 8 elements per thread
  for (int i = threadIdx.x; i < 16 * D; i += 256) {
    int m = i >> 7;
    int k = i & (D - 1);
    int r = row0 + m;
    float v = 0.0f;
    if (r < N) {
      size_t idx = (size_t)r * D + k;
      v = T[idx] + AGG[idx];
    }
    As[m * 132 + k] = v;
  }
  __syncthreads();

  int wave = threadIdx.x >> 5;   // 0..7 -> column block
  int lane = threadIdx.x & 31;
  int col0 = wave * 16;
  int mrow = lane & 15;          // A row owned by this lane
  int n    = lane & 15;          // output column within tile
  int koff = (lane >> 4) << 1;   // 0 for lanes 0-15, 2 for lanes 16-31

  const float* wp = W + col0 + n;  // W[k, col0+n] = wp[k*D]

  v8f c = {};
  #pragma unroll
  for (int kb = 0; kb < D; kb += 4) {
    v2f a;
    a.x = As[mrow * 132 + kb + koff];
    a.y = As[mrow * 132 + kb + koff + 1];
    v2f b;
    b.x = wp[(size_t)(kb + koff) * D];
    b.y = wp[(size_t)(kb + koff + 1) * D];
    // D = A(16x4,f32) * B(4x16,f32) + C ; exact f32 accumulate
    c = __builtin_amdgcn_wmma_f32_16x16x4_f32(
        /*neg_a=*/false, a, /*neg_b=*/false, b,
        /*c_mod=*/(short)0, c, /*reuse_a=*/false, /*reuse_b=*/false);
  }

  // Epilogue: lane l, vgpr v holds (m = v + 8*(l/16), n = l%16)
  int mhalf = (lane >> 4) * 8;
  float bb = bias[col0 + n];
  #pragma unroll
  for (int v = 0; v < 8; ++v) {
    int r = row0 + mhalf + v;
    if (r < N) {
      size_t idx = (size_t)r * D + col0 + n;
      float val = c[v] + bb;
      if (RES) val += RES[idx];
      OUT[idx] = val;
    }
  }
}

// ------------------------------------------------------------ BN statistics
// Block b accumulates rows b, b+NB, ... ; thread = channel. Deterministic.
__global__ __launch_bounds__(D) void dg_bn_stats(
    const float* __restrict__ h, float* __restrict__ partial, int N) {
  int cch = threadIdx.x;
  float s = 0.0f, q = 0.0f;
  for (int r = blockIdx.x; r < N; r += gridDim.x) {
    float v = h[(size_t)r * D + cch];
    s += v;
    q += v * v;
  }
  partial[blockIdx.x * (2 * D) + cch] = s;
  partial[blockIdx.x * (2 * D) + D + cch] = q;
}

// Combine partials serially (deterministic), produce affine A,B:
// h2 = relu(h*A[c] + B[c]) with A = gamma*rsqrt(var+eps), B = beta - mu*A
__global__ __launch_bounds__(D) void dg_bn_finalize(
    const float* __restrict__ partial, int nb,
    const float* __restrict__ gamma, const float* __restrict__ beta,
    float* __restrict__ AB, int N) {
  int cch = threadIdx.x;
  float s = 0.0f, q = 0.0f;
  for (int b = 0; b < nb; ++b) {
    s += partial[b * (2 * D) + cch];
    q += partial[b * (2 * D) + D + cch];
  }
  float invN = 1.0f / (float)N;
  float mu = s * invN;
  float var = q * invN - mu * mu;
  var = fmaxf(var, 0.0f);
  float inv = rsqrtf(var + EPS_BN);
  float A = inv * gamma[cch];
  AB[cch] = A;
  AB[D + cch] = beta[cch] - mu * A;
}

__global__ void dg_bn_apply_relu(const float* __restrict__ h,
                                 const float* __restrict__ AB,
                                 float* __restrict__ out, int total) {
  int i = blockIdx.x * blockDim.x + threadIdx.x;
  int stride = gridDim.x * blockDim.x;
  for (; i < total; i += stride) {
    int cch = i & (D - 1);
    out[i] = fmaxf(fmaf(h[i], AB[cch], AB[D + cch]), 0.0f);
  }
}

// ------------------------------------------------------------------ driver
extern "C" void kernel_launch(void* const* d_in, const int* in_sizes, int n_in,
                              void* d_out, int out_size, void* d_ws, size_t ws_size,
                              hipStream_t stream) {
  const float* x      = (const float*)d_in[0];
  const int*   ei     = (const int*)d_in[1];
  const float* Ws     = (const float*)d_in[2];
  const float* bs     = (const float*)d_in[3];
  const float* gammas = (const float*)d_in[4];
  const float* betas  = (const float*)d_in[5];

  const int N = in_sizes[0] / D;
  const int E = in_sizes[1] / 2;
  const int* src = ei;
  const int* dst = ei + E;

  const size_t ND = (size_t)N * D;
  float* hA      = (float*)d_ws;
  float* hB      = hA + ND;
  float* h2      = hB + ND;
  float* agg     = h2 + ND;
  float* partial = agg + ND;                 // BN_BLOCKS * 2D
  float* AB      = partial + BN_BLOCKS * 2 * D;  // 2D

  const int total   = (int)ND;
  const int zgrid   = 4096;
  const int egrid   = (E + 7) / 8;
  const int ggrid   = (N + 15) / 16;
  const int agrid   = 8192;

  // ---- layer 0: h_A = genconv(x, W0, b0)
  dg_zero<<<zgrid, 256, 0, stream>>>(agg, total);
  dg_scatter_max<<<egrid, 256, 0, stream>>>(x, src, dst, agg, E);
  dg_gemm_wmma<<<ggrid, 256, 0, stream>>>(x, agg, Ws, bs, nullptr, hA, N);

  // ---- layers 1..2: h = genconv(relu(bn(h)), Wl, bl) + h
  const float* prev = hA;
  float* next = hB;
  for (int l = 1; l < 3; ++l) {
    dg_bn_stats<<<BN_BLOCKS, D, 0, stream>>>(prev, partial, N);
    dg_bn_finalize<<<1, D, 0, stream>>>(partial, BN_BLOCKS,
                                        gammas + (size_t)(l - 1) * D,
                                        betas + (size_t)(l - 1) * D, AB, N);
    dg_bn_apply_relu<<<agrid, 256, 0, stream>>>(prev, AB, h2, total);

    dg_zero<<<zgrid, 256, 0, stream>>>(agg, total);
    dg_scatter_max<<<egrid, 256, 0, stream>>>(h2, src, dst, agg, E);
    dg_gemm_wmma<<<ggrid, 256, 0, stream>>>(h2, agg,
                                            Ws + (size_t)l * D * D,
                                            bs + (size_t)l * D,
                                            prev, next, N);
    // ping-pong
    const float* t = prev;
    prev = next;
    next = (float*)t;
  }

  // ---- final: out = relu(bn(h))
  dg_bn_stats<<<BN_BLOCKS, D, 0, stream>>>(prev, partial, N);
  dg_bn_finalize<<<1, D, 0, stream>>>(partial, BN_BLOCKS,
                                      gammas + (size_t)2 * D,
                                      betas + (size_t)2 * D, AB, N);
  dg_bn_apply_relu<<<agrid, 256, 0, stream>>>(prev, AB, (float*)d_out, total);
}